// ImagePrompt_80590766342721
// MI455X (gfx1250) — compile-verified
//
#include <hip/hip_runtime.h>
#include <hip/hip_bf16.h>
#include <math.h>

typedef float v2f __attribute__((ext_vector_type(2)));
typedef float v8f __attribute__((ext_vector_type(8)));

#define B_   128
#define N_   197
#define D_   768
#define P_   30
#define M_   (P_ * 196)   // 5880 rows of im2col / prompt_embed
#define K_   768          // 3*16*16

// ---------------- workspace layout (in floats) ----------------
// xnorm   : 128*768            @ 0
// pnorm   : 30*768             @ 98304
// maxsim  : 128                @ 121344
// idx(int): 128                @ 121472
// A       : 5880*768           @ 121600
// PE      : 5880*768           @ 4637440
// total ≈ 9.16M floats ≈ 36.6 MB

// Kernel 1: per-batch mean over N, then L2-normalize over D.
__global__ void mean_norm_x(const float* __restrict__ x, float* __restrict__ xnorm) {
    __shared__ float red[256];
    int b = blockIdx.x;
    int t = threadIdx.x;
    const float* xb = x + (size_t)b * N_ * D_;
    float s0 = 0.f, s1 = 0.f, s2 = 0.f;
    for (int n = 0; n < N_; ++n) {
        const float* r = xb + (size_t)n * D_ + t;
        s0 += r[0]; s1 += r[256]; s2 += r[512];
    }
    const float invN = 1.0f / 197.0f;
    s0 *= invN; s1 *= invN; s2 *= invN;
    red[t] = s0 * s0 + s1 * s1 + s2 * s2;
    __syncthreads();
    for (int off = 128; off > 0; off >>= 1) {
        if (t < off) red[t] += red[t + off];
        __syncthreads();
    }
    float inv = 1.0f / sqrtf(fmaxf(red[0], 1e-12f));
    xnorm[(size_t)b * D_ + t]       = s0 * inv;
    xnorm[(size_t)b * D_ + t + 256] = s1 * inv;
    xnorm[(size_t)b * D_ + t + 512] = s2 * inv;
}

// Kernel 2: L2-normalize prompt_key rows.
__global__ void norm_pkey(const float* __restrict__ pk, float* __restrict__ pnorm) {
    __shared__ float red[256];
    int p = blockIdx.x;
    int t = threadIdx.x;
    const float* r = pk + (size_t)p * D_;
    float v0 = r[t], v1 = r[t + 256], v2 = r[t + 512];
    red[t] = v0 * v0 + v1 * v1 + v2 * v2;
    __syncthreads();
    for (int off = 128; off > 0; off >>= 1) {
        if (t < off) red[t] += red[t + off];
        __syncthreads();
    }
    float inv = 1.0f / sqrtf(fmaxf(red[0], 1e-12f));
    pnorm[(size_t)p * D_ + t]       = v0 * inv;
    pnorm[(size_t)p * D_ + t + 256] = v1 * inv;
    pnorm[(size_t)p * D_ + t + 512] = v2 * inv;
}

// Kernel 3: similarity row + wave32 argmax. One wave per batch element.
__global__ void sim_argmax(const float* __restrict__ xnorm, const float* __restrict__ pnorm,
                           float* __restrict__ sim_out, float* __restrict__ idx_out_f,
                           int* __restrict__ idx_ws, float* __restrict__ maxsim_ws) {
    int b = blockIdx.x;
    int t = threadIdx.x;  // 0..31, wave32
    float v = -1e30f;
    if (t < P_) {
        const float* xr = xnorm + (size_t)b * D_;
        const float* pr = pnorm + (size_t)t * D_;
        float s = 0.f;
        for (int d = 0; d < D_; ++d) s += xr[d] * pr[d];
        sim_out[b * P_ + t] = s;
        v = s;
    }
    int vi = t;
    for (int off = 16; off > 0; off >>= 1) {
        float ov = __shfl_down(v, off, 32);
        int   oi = __shfl_down(vi, off, 32);
        if (ov > v || (ov == v && oi < vi)) { v = ov; vi = oi; }
    }
    if (t == 0) {
        idx_ws[b] = vi;
        idx_out_f[b] = (float)vi;
        maxsim_ws[b] = v;
    }
}

// Kernel 4: reduce_sim = sum_b maxsim[b] / B  (deterministic single-block reduce).
__global__ void reduce_sim_k(const float* __restrict__ maxsim, float* __restrict__ out_scalar) {
    __shared__ float red[128];
    int t = threadIdx.x;
    red[t] = maxsim[t];
    __syncthreads();
    for (int off = 64; off > 0; off >>= 1) {
        if (t < off) red[t] += red[t + off];
        __syncthreads();
    }
    if (t == 0) out_scalar[0] = red[0] * (1.0f / 128.0f);
}

// Kernel 5: fused bilinear resize (32->224, half-pixel centers) + im2col.
// A[m, k], m = p*196 + py*14 + px, k = c*256 + ky*16 + kx.
__global__ void resize_im2col(const float* __restrict__ prompt, float* __restrict__ A) {
    int i = blockIdx.x * blockDim.x + threadIdx.x;
    if (i >= M_ * K_) return;
    int k = i % K_;
    int m = i / K_;
    int kx = k & 15, ky = (k >> 4) & 15, c = k >> 8;
    int px = m % 14; int tt = m / 14; int py = tt % 14; int p = tt / 14;
    int ox = px * 16 + kx, oy = py * 16 + ky;
    float fx = (ox + 0.5f) * (1.0f / 7.0f) - 0.5f;
    float fy = (oy + 0.5f) * (1.0f / 7.0f) - 0.5f;
    float flx = floorf(fx), fly = floorf(fy);
    float wx = fx - flx, wy = fy - fly;
    int x0 = (int)flx, y0 = (int)fly;
    int x1 = x0 + 1, y1 = y0 + 1;
    x0 = min(max(x0, 0), 31); x1 = min(max(x1, 0), 31);
    y0 = min(max(y0, 0), 31); y1 = min(max(y1, 0), 31);
    const float* img = prompt + ((size_t)(p * 3 + c) * 32 * 32);
    float v00 = img[y0 * 32 + x0], v01 = img[y0 * 32 + x1];
    float v10 = img[y1 * 32 + x0], v11 = img[y1 * 32 + x1];
    float v = (1.f - wy) * ((1.f - wx) * v00 + wx * v01)
            +         wy * ((1.f - wx) * v10 + wx * v11);
    A[i] = v;
}

// Kernel 6: PE[m, d] = sum_k A[m,k] * W[d,k] + bias[d] via V_WMMA_F32_16X16X4_F32.
// One wave per 16(M) x 64(N) output tile; A-fragment reused across 4 accumulators.
__global__ void __launch_bounds__(32) conv_gemm_wmma(const float* __restrict__ A,
                                                     const float* __restrict__ W,
                                                     const float* __restrict__ bias,
                                                     float* __restrict__ PE) {
    int l   = threadIdx.x;
    int l16 = l & 15;
    int hi  = l >> 4;
    int tileM = blockIdx.x * 16;
    int tileN = blockIdx.y * 64;
    int rowA = tileM + l16; if (rowA >= M_) rowA = M_ - 1;

    const float* Ar = A + (size_t)rowA * K_ + 2 * hi;
    const float* W0 = W + (size_t)(tileN +  0 + l16) * K_ + 2 * hi;
    const float* W1 = W0 + 16 * K_;
    const float* W2 = W0 + 32 * K_;
    const float* W3 = W0 + 48 * K_;

    v8f acc[4];
    acc[0] = (v8f)(0.0f); acc[1] = (v8f)(0.0f);
    acc[2] = (v8f)(0.0f); acc[3] = (v8f)(0.0f);

    for (int k0 = 0; k0 < K_; k0 += 4) {
        v2f a  = *(const v2f*)(Ar + k0);
        v2f b0 = *(const v2f*)(W0 + k0);
        v2f b1 = *(const v2f*)(W1 + k0);
        v2f b2 = *(const v2f*)(W2 + k0);
        v2f b3 = *(const v2f*)(W3 + k0);
        acc[0] = __builtin_amdgcn_wmma_f32_16x16x4_f32(false, a, false, b0, (short)0, acc[0], false, false);
        acc[1] = __builtin_amdgcn_wmma_f32_16x16x4_f32(false, a, false, b1, (short)0, acc[1], false, false);
        acc[2] = __builtin_amdgcn_wmma_f32_16x16x4_f32(false, a, false, b2, (short)0, acc[2], false, false);
        acc[3] = __builtin_amdgcn_wmma_f32_16x16x4_f32(false, a, false, b3, (short)0, acc[3], false, false);
    }

#pragma unroll
    for (int nt = 0; nt < 4; ++nt) {
        int col = tileN + nt * 16 + l16;
        float bs = bias[col];
        const float* cf = (const float*)&acc[nt];
#pragma unroll
        for (int r = 0; r < 8; ++r) {
            int row = tileM + r + 8 * hi;   // C/D layout: VGPR r -> M=r (+8 for upper lane half)
            if (row < M_) PE[(size_t)row * D_ + col] = cf[r] + bs;
        }
    }
}

// Kernel 7: gather prompt_embed by idx and concat with x_embed -> prompted_embedding.
__global__ void assemble_out(const float* __restrict__ x, const float* __restrict__ PE,
                             const int* __restrict__ idx, float* __restrict__ out) {
    const int ROW4 = D_ / 4;                       // 192 float4 per row
    const long total = (long)B_ * 393 * ROW4;      // 9,658,368
    long i = (long)blockIdx.x * blockDim.x + threadIdx.x;
    long stride = (long)gridDim.x * blockDim.x;
    for (; i < total; i += stride) {
        int q = (int)(i % ROW4);
        long t = i / ROW4;
        int n = (int)(t % 393);
        int b = (int)(t / 393);
        float4 v;
        if (n < 196) {
            int p = idx[b];
            v = *(const float4*)(PE + ((size_t)(p * 196 + n) * D_ + q * 4));
        } else {
            v = *(const float4*)(x + ((size_t)(b * N_ + (n - 196)) * D_ + q * 4));
        }
        *(float4*)(out + (((size_t)b * 393 + n) * D_ + q * 4)) = v;
    }
}

extern "C" void kernel_launch(void* const* d_in, const int* in_sizes, int n_in,
                              void* d_out, int out_size, void* d_ws, size_t ws_size,
                              hipStream_t stream) {
    (void)in_sizes; (void)n_in; (void)out_size; (void)ws_size;
    const float* x_embed    = (const float*)d_in[0];
    const float* prompt     = (const float*)d_in[1];
    const float* prompt_key = (const float*)d_in[2];
    const float* patch_w    = (const float*)d_in[3];
    const float* patch_b    = (const float*)d_in[4];
    float* out = (float*)d_out;
    float* ws  = (float*)d_ws;

    float* xnorm  = ws;
    float* pnorm  = ws + 98304;
    float* maxsim = ws + 121344;
    int*   idxws  = (int*)(ws + 121472);
    float* A      = ws + 121600;
    float* PE     = ws + 4637440;

    float* out_scalar = out + 38633472;           // reduce_sim
    float* out_sim    = out + 38633473;           // similarity [128,30]
    float* out_idx    = out + 38637313;           // idx [128] (as float)

    mean_norm_x<<<B_, 256, 0, stream>>>(x_embed, xnorm);
    norm_pkey<<<P_, 256, 0, stream>>>(prompt_key, pnorm);
    sim_argmax<<<B_, 32, 0, stream>>>(xnorm, pnorm, out_sim, out_idx, idxws, maxsim);
    reduce_sim_k<<<1, 128, 0, stream>>>(maxsim, out_scalar);
    resize_im2col<<<(M_ * K_ + 255) / 256, 256, 0, stream>>>(prompt, A);
    dim3 g6((M_ + 15) / 16, D_ / 64);
    conv_gemm_wmma<<<g6, 32, 0, stream>>>(A, patch_w, patch_b, PE);
    assemble_out<<<4096, 256, 0, stream>>>(x_embed, PE, idxws, out);
}